// DemandMap_33921651704719
// MI455X (gfx1250) — compile-verified
//
#include <hip/hip_runtime.h>

// CDNA5 / gfx1250: wave32, WMMA f32 16x16x4 used for the exact 4-site f32 reduction.

typedef __attribute__((ext_vector_type(2))) float v2f;
typedef __attribute__((ext_vector_type(8))) float v8f;

namespace {
constexpr int NBX = 2048;
constexpr int NBY = 2048;
constexpr int HEIGHT = 4096;              // site_type_map is [4096][4096], idx = x*4096 + y
constexpr long long NBINS = (long long)NBX * NBY;     // 4194304 bins per map
constexpr int NGROUPS = (int)(NBINS / 16);            // 262144 groups of 16 bins
constexpr int WAVES_PER_BLOCK = 8;                    // 256 threads = 8 wave32
constexpr int GROUPS_PER_WAVE = 4;
constexpr int NBLOCKS = NGROUPS / (WAVES_PER_BLOCK * GROUPS_PER_WAVE);  // 8192
}

__global__ __launch_bounds__(256) void demand_map_wmma_kernel(
    const int* __restrict__ st, const float* __restrict__ nsx,
    const float* __restrict__ nsy, float* __restrict__ out)
{
  const int lane = threadIdx.x & 31;
  const int wave = threadIdx.x >> 5;
  const int n    = lane & 15;   // bin column within group == WMMA N index
  const int half = lane >> 4;   // which pair of K-slots (sites) this lane feeds

  // Uniform 4-entry LUTs -> scalar loads (index 0 never contributes).
  const float s1x = nsx[1], s2x = nsx[2], s3x = nsx[3];
  const float s1y = nsy[1], s2y = nsy[2], s3y = nsy[3];

  const v2f aOnes = {1.0f, 1.0f};   // A[16x4] = ones (layout-independent)
  const v8f cZero = {};             // C = 0

  const int gBase = (blockIdx.x * WAVES_PER_BLOCK + wave) * GROUPS_PER_WAVE;

  for (int gi = 0; gi < GROUPS_PER_WAVE; ++gi) {
    const int g  = gBase + gi;
    const int b  = (g << 4) + n;        // global bin index for this lane-column
    const int bi = b >> 11;             // x-bin (128 groups per bin row -> no straddle)
    const int bj = b & (NBY - 1);       // y-bin
    const int x  = 2 * bi + half;       // site row this lane covers
    const int y  = 2 * bj;              // even -> int2 load is 8B aligned

    const int2 tt = *reinterpret_cast<const int2*>(st + (size_t)x * HEIGHT + y);

    const float xf = (float)x;

    // Site 0: (x, y). Replicate reference arithmetic: wx = fl(x+nw) - x, etc.
    float b00, b01, b02;
    {
      const int   t  = tt.x;
      const float nw = (t == 1) ? s1x : ((t == 2) ? s2x : s3x);
      const float nh = (t == 1) ? s1y : ((t == 2) ? s2y : s3y);
      const float yf = (float)y;
      const float wx = (xf + nw) - xf;
      const float wy = (yf + nh) - yf;
      const float a  = wx * wy;
      b00 = (t == 1) ? a : 0.0f;   // slot 0 (type 1)
      b01 = (t == 2) ? a : 0.0f;   // slot 1 (type 2)
      b02 = (t == 3) ? a : 0.0f;   // slot 2 (type 3)
    }
    // Site 1: (x, y+1)
    float b10, b11, b12;
    {
      const int   t  = tt.y;
      const float nw = (t == 1) ? s1x : ((t == 2) ? s2x : s3x);
      const float nh = (t == 1) ? s1y : ((t == 2) ? s2y : s3y);
      const float yf = (float)(y + 1);
      const float wx = (xf + nw) - xf;
      const float wy = (yf + nh) - yf;
      const float a  = wx * wy;
      b10 = (t == 1) ? a : 0.0f;
      b11 = (t == 2) ? a : 0.0f;
      b12 = (t == 3) ? a : 0.0f;
    }

    // B[4x16] per slot: lane (n, n+16) supplies column n's four K entries.
    const v2f B0 = {b00, b10};
    const v2f B1 = {b01, b11};
    const v2f B2 = {b02, b12};

    // D[m][n] = sum_k B_s[k][n]  (A = ones) -> every lane's d[0] == c_s(bin n).
    // Full-precision f32 WMMA: exact 4-term f32 accumulation, matches reference dtype.
    v8f d0 = __builtin_amdgcn_wmma_f32_16x16x4_f32(false, aOnes, false, B0,
                                                   (short)0, cZero, false, false);
    v8f d1 = __builtin_amdgcn_wmma_f32_16x16x4_f32(false, aOnes, false, B1,
                                                   (short)0, cZero, false, false);
    v8f d2 = __builtin_amdgcn_wmma_f32_16x16x4_f32(false, aOnes, false, B2,
                                                   (short)0, cZero, false, false);

    const float c0 = d0[0];   // type-1 demand for bin b
    const float c5 = d1[0];   // type-2
    const float c6 = d2[0];   // type-3

    const float v0 = 4.0f - c0;                     // bin_area = 2*2
    const float v1 = half ? (4.0f - c5) : v0;
    const float v2 = half ? (4.0f - c6) : v0;

    // Branchless stores: lower half -> maps 0,1,2 ; upper half -> maps 4,5,6 ;
    // both halves -> map 3 (identical value). EXEC stays all-ones (WMMA req).
    const size_t ob   = (size_t)b;
    const size_t mofs = (size_t)half * 4u * (size_t)NBINS;
    out[mofs + ob]                       = v0;   // map 0 or 4
    out[mofs + (size_t)NBINS + ob]       = v1;   // map 1 or 5
    out[mofs + 2 * (size_t)NBINS + ob]   = v2;   // map 2 or 6
    out[3 * (size_t)NBINS + ob]          = v0;   // map 3
  }
}

extern "C" void kernel_launch(void* const* d_in, const int* in_sizes, int n_in,
                              void* d_out, int out_size, void* d_ws, size_t ws_size,
                              hipStream_t stream) {
  const int*   st  = (const int*)d_in[0];    // site_type_map [4096*4096] int32
  const float* nsx = (const float*)d_in[1];  // node_size_x [4] f32
  const float* nsy = (const float*)d_in[2];  // node_size_y [4] f32
  float*       out = (float*)d_out;          // [7, 2048, 2048] f32

  demand_map_wmma_kernel<<<NBLOCKS, 256, 0, stream>>>(st, nsx, nsy, out);
}